// GraphSeqGenerator_85143431675937
// MI455X (gfx1250) — compile-verified
//
#include <hip/hip_runtime.h>
#include <math.h>

// CDNA5 (gfx1250) wave32 WMMA fragments for V_WMMA_F32_16X16X4_F32:
//   A (16x4 f32)  -> 2 VGPRs/lane (v2f): lanes 0-15 K{0,1}, lanes 16-31 K{2,3}
//   B (4x16 f32)  -> 2 VGPRs/lane (v2f): same K split, N = lane&15
//   C/D (16x16)   -> 8 VGPRs/lane (v8f): VGPR j = rows {j, j+8}, N = lane&15
typedef __attribute__((ext_vector_type(2))) float v2f;
typedef __attribute__((ext_vector_type(8))) float v8f;

// ---------------------------------------------------------------------------
// Elementwise utility kernels
// ---------------------------------------------------------------------------
__global__ void k_fill0(float* __restrict__ p, long n) {
  long i = (long)blockIdx.x * blockDim.x + threadIdx.x;
  if (i < n) p[i] = 0.0f;
}

__global__ void k_copy(float* __restrict__ d, const float* __restrict__ s, long n) {
  long i = (long)blockIdx.x * blockDim.x + threadIdx.x;
  if (i < n) d[i] = s[i];
}

__global__ void k_mul(float* __restrict__ o, const float* __restrict__ a,
                      const float* __restrict__ b, long n) {
  long i = (long)blockIdx.x * blockDim.x + threadIdx.x;
  if (i < n) o[i] = a[i] * b[i];
}

// ---------------------------------------------------------------------------
// Scaled-Laplacian weights:  lw = -d^-1/2[src] * ew * d^-1/2[dst], self-loops 0
// ---------------------------------------------------------------------------
__global__ void k_deg(float* __restrict__ deg, const int* __restrict__ src,
                      const int* __restrict__ dst, const float* __restrict__ ew, int E) {
  int e = blockIdx.x * blockDim.x + threadIdx.x;
  if (e >= E) return;
  int s = src[e];
  float w = (s == dst[e]) ? 0.0f : ew[e];
  if (w != 0.0f) atomicAdd(&deg[s], w);
}

__global__ void k_dis(float* __restrict__ dis, const float* __restrict__ deg, int N) {
  int i = blockIdx.x * blockDim.x + threadIdx.x;
  if (i >= N) return;
  float d = deg[i];
  dis[i] = (d > 0.0f) ? rsqrtf(fmaxf(d, 1e-30f)) : 0.0f;
}

__global__ void k_lw(float* __restrict__ lw, const float* __restrict__ dis,
                     const int* __restrict__ src, const int* __restrict__ dst,
                     const float* __restrict__ ew, int E) {
  int e = blockIdx.x * blockDim.x + threadIdx.x;
  if (e >= E) return;
  int s = src[e], d = dst[e];
  float w = (s == d) ? 0.0f : ew[e];
  lw[e] = -dis[s] * w * dis[d];
}

// ---------------------------------------------------------------------------
// SpMV scatter: out[dst, :] += lw[e] * v[src, :], compile-time C.
// Each thread handles min(C,16) channels so the src/dst/lw index loads are
// amortized; atomics resolve in L2 (working set < 192 MB L2, HBM stays idle).
// ---------------------------------------------------------------------------
template <int C>
__global__ void k_spmv(float* __restrict__ out, const float* __restrict__ v,
                       const int* __restrict__ src, const int* __restrict__ dst,
                       const float* __restrict__ lw, int E) {
  constexpr int CPT = (C >= 16) ? 16 : C;   // channels per thread (multiple of 4)
  constexpr int TPE = C / CPT;              // threads per edge
  long i = (long)blockIdx.x * blockDim.x + threadIdx.x;
  long total = (long)E * TPE;
  if (i >= total) return;
  int e  = (int)(i / TPE);
  int c0 = ((int)(i % TPE)) * CPT;
  float w = lw[e];
  if (w == 0.0f) return;
  const float* vp = v + (long)src[e] * C + c0;
  float* op = out + (long)dst[e] * C + c0;
#pragma unroll
  for (int c = 0; c < CPT; c += 4) {
    const float4 val = *(const float4*)(vp + c);
    atomicAdd(op + c + 0, w * val.x);
    atomicAdd(op + c + 1, w * val.y);
    atomicAdd(op + c + 2, w * val.z);
    atomicAdd(op + c + 3, w * val.w);
  }
}

// ---------------------------------------------------------------------------
// Multi-term fp32 WMMA GEMM (compile-time K per term and column-tile count,
// fully unrolled):  Out = epi( sum_i A_i @ W_i + b0 + b1 )
// Each wave owns a 16-row strip and computes ALL COLT 16-column tiles at once
// (COLT v8f accumulators), so each A fragment is loaded exactly once and feeds
// COLT back-to-back WMMAs. The full K x (COLT*16) weight slab of every term is
// staged in LDS per block (vectorized b128 stores, zero-padded past Cout).
// Assumes Nrows % 16 == 0 (N = 50000 = 3125 * 16).
// ---------------------------------------------------------------------------
template <int KT, int COLT>
__device__ __forceinline__ void stage_w(float* __restrict__ sw,
                                        const float* __restrict__ W,
                                        int Cout, int tid, int nthr) {
  constexpr int CW = COLT * 16;
  if (Cout == CW) {                          // exact slab: float4 staging
    const float4* W4 = (const float4*)W;
    float4* s4 = (float4*)sw;
    for (int idx = tid; idx < KT * CW / 4; idx += nthr) s4[idx] = W4[idx];
  } else {                                   // ragged (Cout=4): zero-pad
    for (int idx = tid; idx < KT * CW; idx += nthr) {
      int row = idx / CW;
      int col = idx % CW;
      sw[idx] = (col < Cout) ? W[(long)row * Cout + col] : 0.0f;
    }
  }
}

template <int KT, int RELUA, int COLT>
__device__ __forceinline__ void wmma_term(v8f (&acc)[COLT],
                                          const float* __restrict__ A,
                                          const float* __restrict__ sw,
                                          int r0, int lane) {
  constexpr int CW = COLT * 16;
  const int m   = lane & 15;
  const int kof = (lane >> 4) << 1;   // 0 or 2
  const int n   = lane & 15;
  const float2* arow = (const float2*)(A + (long)(r0 + m) * KT);
#pragma unroll
  for (int k0 = 0; k0 < KT; k0 += 4) {
    float2 av = arow[(k0 + kof) >> 1];            // one global_load_b64
    v2f a;
    a.x = RELUA ? fmaxf(av.x, 0.0f) : av.x;
    a.y = RELUA ? fmaxf(av.y, 0.0f) : av.y;
#pragma unroll
    for (int ct = 0; ct < COLT; ++ct) {           // COLT WMMAs per A fragment
      v2f b;
      b.x = sw[(k0 + kof) * CW + ct * 16 + n];    // ds_load, shared by 8 waves
      b.y = sw[(k0 + kof + 1) * CW + ct * 16 + n];
      acc[ct] = __builtin_amdgcn_wmma_f32_16x16x4_f32(false, a, false, b,
                                                      (short)0, acc[ct],
                                                      false, false);
    }
  }
}

template <int KT0, int KT1, int KT2, int KT3, int EPI, int RELUA, int COLT>
__global__ __launch_bounds__(256) void k_wmma_gemm(
    float* __restrict__ Out, int Nrows, int Cout,
    const float* A0, const float* W0, const float* A1, const float* W1,
    const float* A2, const float* W2, const float* A3, const float* W3,
    const float* b0, const float* b1, const float* Zg, const float* Hg) {
  constexpr int KTOT = KT0 + KT1 + KT2 + KT3;
  constexpr int CW = COLT * 16;
  __shared__ float sw[KTOT * CW];

  const int tid = threadIdx.x;
  stage_w<KT0, COLT>(sw, W0, Cout, tid, blockDim.x);
  if (KT1 > 0) stage_w<KT1, COLT>(sw + KT0 * CW, W1, Cout, tid, blockDim.x);
  if (KT2 > 0) stage_w<KT2, COLT>(sw + (KT0 + KT1) * CW, W2, Cout, tid, blockDim.x);
  if (KT3 > 0) stage_w<KT3, COLT>(sw + (KT0 + KT1 + KT2) * CW, W3, Cout, tid, blockDim.x);
  __syncthreads();

  const int lane = tid & 31;
  const int rowTile = blockIdx.x * (blockDim.x >> 5) + (tid >> 5);
  const int r0 = rowTile << 4;
  if (r0 >= Nrows) return;   // wave-uniform; EXEC all-1s for the WMMA chain

  v8f acc[COLT] = {};
  wmma_term<KT0, RELUA, COLT>(acc, A0, sw, r0, lane);
  if (KT1 > 0) wmma_term<KT1, RELUA, COLT>(acc, A1, sw + KT0 * CW, r0, lane);
  if (KT2 > 0) wmma_term<KT2, RELUA, COLT>(acc, A2, sw + (KT0 + KT1) * CW, r0, lane);
  if (KT3 > 0) wmma_term<KT3, RELUA, COLT>(acc, A3, sw + (KT0 + KT1 + KT2) * CW, r0, lane);

  const int nn = lane & 15;
  const int rbase = r0 + ((lane >> 4) << 3);
#pragma unroll
  for (int ct = 0; ct < COLT; ++ct) {
    int col = ct * 16 + nn;
    if (col < Cout) {                      // per-lane guard; ragged only Cout=4
      float bb = 0.0f;
      if (b0) bb += b0[col];
      if (b1) bb += b1[col];
#pragma unroll
      for (int j = 0; j < 8; ++j) {
        int row = rbase + j;
        float vv = acc[ct][j] + bb;
        if (EPI == 1) {
          vv = 1.0f / (1.0f + expf(-vv));        // sigmoid gate
        } else if (EPI == 2) {
          float z = Zg[(long)row * Cout + col];
          float h = Hg[(long)row * Cout + col];
          vv = z * h + (1.0f - z) * tanhf(vv);   // fused GRU state update
        }
        Out[(long)row * Cout + col] = vv;
      }
    }
  }
}

// ---------------------------------------------------------------------------
// Host-side helpers (all launches on `stream`; graph-capture safe)
// ---------------------------------------------------------------------------
static inline int ceil_div_l(long a, long b) { return (int)((a + b - 1) / b); }

static void fill0(hipStream_t st, float* p, long n) {
  k_fill0<<<ceil_div_l(n, 256), 256, 0, st>>>(p, n);
}
static void copyf(hipStream_t st, float* d, const float* s, long n) {
  k_copy<<<ceil_div_l(n, 256), 256, 0, st>>>(d, s, n);
}
static void spmv(hipStream_t st, float* out, const float* v, const int* src,
                 const int* dst, const float* lw, int E, int C) {
  if (C == 4) {
    k_spmv<4><<<ceil_div_l((long)E, 256), 256, 0, st>>>(out, v, src, dst, lw, E);
  } else if (C == 32) {
    k_spmv<32><<<ceil_div_l((long)E * 2, 256), 256, 0, st>>>(out, v, src, dst, lw, E);
  } else {
    k_spmv<64><<<ceil_div_l((long)E * 4, 256), 256, 0, st>>>(out, v, src, dst, lw, E);
  }
}

// Fused 4-term gate/candidate GEMM for the ChebGRU (Cout = 64 -> COLT = 4).
static void gemm_gru(hipStream_t st, float* Out, int N, int cinX,
                     const float* X, const float* Wx0, const float* LX, const float* Wx1,
                     const float* H, const float* Wh0, const float* LH, const float* Wh1,
                     const float* bx, const float* bh,
                     const float* Zg, const float* Hg, int epi) {
  dim3 grd(ceil_div_l(ceil_div_l(N, 16), 8), 1);
  if (cinX == 4) {
    if (epi == 1)
      k_wmma_gemm<4, 4, 64, 64, 1, 0, 4><<<grd, 256, 0, st>>>(
          Out, N, 64, X, Wx0, LX, Wx1, H, Wh0, LH, Wh1, bx, bh, nullptr, nullptr);
    else
      k_wmma_gemm<4, 4, 64, 64, 2, 0, 4><<<grd, 256, 0, st>>>(
          Out, N, 64, X, Wx0, LX, Wx1, H, Wh0, LH, Wh1, bx, bh, Zg, Hg);
  } else {  // cinX == 32
    if (epi == 1)
      k_wmma_gemm<32, 32, 64, 64, 1, 0, 4><<<grd, 256, 0, st>>>(
          Out, N, 64, X, Wx0, LX, Wx1, H, Wh0, LH, Wh1, bx, bh, nullptr, nullptr);
    else
      k_wmma_gemm<32, 32, 64, 64, 2, 0, 4><<<grd, 256, 0, st>>>(
          Out, N, 64, X, Wx0, LX, Wx1, H, Wh0, LH, Wh1, bx, bh, Zg, Hg);
  }
}

// Out = relu(A) @ W + b, A: N x 64, Cout in {32, 4}
static void gemm_lin(hipStream_t st, float* Out, int N, int Cout,
                     const float* A, const float* W, const float* b) {
  dim3 grd(ceil_div_l(ceil_div_l(N, 16), 8), 1);
  if (Cout == 32)
    k_wmma_gemm<64, 0, 0, 0, 0, 1, 2><<<grd, 256, 0, st>>>(
        Out, N, Cout, A, W, nullptr, nullptr, nullptr, nullptr, nullptr, nullptr,
        b, nullptr, nullptr, nullptr);
  else
    k_wmma_gemm<64, 0, 0, 0, 0, 1, 1><<<grd, 256, 0, st>>>(
        Out, N, Cout, A, W, nullptr, nullptr, nullptr, nullptr, nullptr, nullptr,
        b, nullptr, nullptr, nullptr);
}

struct ChebP { const float* W0; const float* W1; const float* b; };

// One ChebConv(K=2)-GRU cell step; H updated in place (epilogue is elementwise
// per (row,col): each storing thread reads H/Z at exactly the address it writes).
static void gru_step(hipStream_t st, int N, int E,
                     const int* src, const int* dst, const float* lw,
                     const float* X, int cinX, float* H, int ch,
                     ChebP xz, ChebP hz, ChebP xr, ChebP hr, ChebP xh, ChebP hh,
                     float* LX, float* LH, float* Zb, float* Rb,
                     float* HRb, float* LHR) {
  fill0(st, LX, (long)N * cinX);
  spmv(st, LX, X, src, dst, lw, E, cinX);
  fill0(st, LH, (long)N * ch);
  spmv(st, LH, H, src, dst, lw, E, ch);
  // Z = sigmoid(X@Wxz0 + LX@Wxz1 + H@Whz0 + LH@Whz1 + bxz + bhz)
  gemm_gru(st, Zb, N, cinX, X, xz.W0, LX, xz.W1, H, hz.W0, LH, hz.W1,
           xz.b, hz.b, nullptr, nullptr, 1);
  // R = sigmoid(...)
  gemm_gru(st, Rb, N, cinX, X, xr.W0, LX, xr.W1, H, hr.W0, LH, hr.W1,
           xr.b, hr.b, nullptr, nullptr, 1);
  k_mul<<<ceil_div_l((long)N * ch, 256), 256, 0, st>>>(HRb, H, Rb, (long)N * ch);
  fill0(st, LHR, (long)N * ch);
  spmv(st, LHR, HRb, src, dst, lw, E, ch);
  // H = Z*H + (1-Z)*tanh(X@Wxh0 + LX@Wxh1 + HR@Whh0 + LHR@Whh1 + bxh + bhh)
  gemm_gru(st, H, N, cinX, X, xh.W0, LX, xh.W1, HRb, hh.W0, LHR, hh.W1,
           xh.b, hh.b, Zb, H, 2);
}

// ---------------------------------------------------------------------------
// kernel_launch
// Input order: x, edge_index, edge_weight, h_enc, h_dec, then params leaves in
// JAX pytree (alphabetical) order:
//   dec_gru{hh,hr,hz,xh,xr,xz}x{W,b} -> 5..16, dec_lin{W,b} -> 17,18,
//   enc_gru{...} -> 19..30, enc_lin{W,b} -> 31,32
// ---------------------------------------------------------------------------
extern "C" void kernel_launch(void* const* d_in, const int* in_sizes, int n_in,
                              void* d_out, int out_size, void* d_ws, size_t ws_size,
                              hipStream_t stream) {
  const int HDIM = 64, FIN = 4, LATD = 32;
  const float* x  = (const float*)d_in[0];
  const int*   ei = (const int*)d_in[1];
  const float* ew = (const float*)d_in[2];
  const float* h_enc_in = (const float*)d_in[3];
  const float* h_dec_in = (const float*)d_in[4];
  const int E = in_sizes[2];
  const int N = in_sizes[3] / HDIM;
  const int T = in_sizes[0] / (N * FIN);
  const int* src = ei;
  const int* dst = ei + E;

  auto P = [&](int i) { return (const float*)d_in[i]; };
  auto cheb = [&](int wi, int bi, int cin, int cout) -> ChebP {
    const float* W = P(wi);
    return ChebP{W, W + (size_t)cin * cout, P(bi)};
  };

  // Workspace layout (floats); everything fits comfortably in the 192 MB L2.
  float* ws  = (float*)d_ws;
  float* deg = ws;
  float* dis = deg + N;
  float* lw  = dis + N;
  float* LX  = lw + E;
  float* LH  = LX + (long)N * HDIM;
  float* Zb  = LH + (long)N * HDIM;
  float* Rb  = Zb + (long)N * HDIM;
  float* HRb = Rb + (long)N * HDIM;
  float* LHR = HRb + (long)N * HDIM;
  float* He  = LHR + (long)N * HDIM;
  float* Hd  = He + (long)N * HDIM;
  float* zl  = Hd + (long)N * HDIM;

  // Laplacian normalization (once per call).
  fill0(stream, deg, N);
  k_deg<<<ceil_div_l(E, 256), 256, 0, stream>>>(deg, src, dst, ew, E);
  k_dis<<<ceil_div_l(N, 256), 256, 0, stream>>>(dis, deg, N);
  k_lw<<<ceil_div_l(E, 256), 256, 0, stream>>>(lw, dis, src, dst, ew, E);

  // Initial hidden states (copied so inputs are never mutated).
  copyf(stream, He, h_enc_in, (long)N * HDIM);
  copyf(stream, Hd, h_dec_in, (long)N * HDIM);

  ChebP dhh = cheb(5, 6, HDIM, HDIM),   dhr = cheb(7, 8, HDIM, HDIM);
  ChebP dhz = cheb(9, 10, HDIM, HDIM),  dxh = cheb(11, 12, LATD, HDIM);
  ChebP dxr = cheb(13, 14, LATD, HDIM), dxz = cheb(15, 16, LATD, HDIM);
  const float* dlW = P(17); const float* dlb = P(18);
  ChebP ehh = cheb(19, 20, HDIM, HDIM), ehr = cheb(21, 22, HDIM, HDIM);
  ChebP ehz = cheb(23, 24, HDIM, HDIM), exh = cheb(25, 26, FIN, HDIM);
  ChebP exr = cheb(27, 28, FIN, HDIM),  exz = cheb(29, 30, FIN, HDIM);
  const float* elW = P(31); const float* elb = P(32);

  for (int t = 0; t < T; ++t) {
    const float* Xt = x + (long)t * N * FIN;
    // Encoder ChebGRU
    gru_step(stream, N, E, src, dst, lw, Xt, FIN, He, HDIM,
             exz, ehz, exr, ehr, exh, ehh, LX, LH, Zb, Rb, HRb, LHR);
    // z = relu(He) @ Wenc + benc
    gemm_lin(stream, zl, N, LATD, He, elW, elb);
    // Decoder ChebGRU
    gru_step(stream, N, E, src, dst, lw, zl, LATD, Hd, HDIM,
             dxz, dhz, dxr, dhr, dxh, dhh, LX, LH, Zb, Rb, HRb, LHR);
  }
  // Only last step's prediction is returned: y = relu(Hd) @ Wdec + bdec
  gemm_lin(stream, (float*)d_out, N, FIN, Hd, dlW, dlb);
}